// VanRossum_22016002359702
// MI455X (gfx1250) — compile-verified
//
#include <hip/hip_runtime.h>
#include <math.h>

// Van Rossum distance, MI455X (gfx1250, wave32).
// Per-chunk causal scan as D = L*X via chained V_WMMA_F32_16X16X4_F32;
// cross-segment carries via ds_bpermute Kogge-Stone scan; rows split into
// 4 independent chunks (d^4096 == 0 in f32) combined algebraically:
//   sum(y_true^2) = alpha + 2*c*beta + c^2*gamma,  c = prev chunk end value.

typedef float v2f __attribute__((ext_vector_type(2)));
typedef float v8f __attribute__((ext_vector_type(8)));

#define LEN        16384
#define ROWS       1024
#define TAUF       20.0f
#define CHUNKS     4                    // per row; d^(LEN/CHUNKS) == 0 in f32
#define CHUNK_LEN  (LEN / CHUNKS)       // 4096
#define NCHUNK     (ROWS * CHUNKS)      // 4096 independent waves
#define TILES      (CHUNK_LEN / 256)    // 16 tiles of 256 elements
#define BETA_TILES 8                    // d^(256*t) == 0 for t >= 8

__device__ __forceinline__ float bcastf(float v, int srcLane) {
    return __int_as_float(
        __builtin_amdgcn_ds_bpermute(srcLane << 2, __float_as_int(v)));
}

__device__ __forceinline__ float dpow(int p) {   // decay^p, p >= 0
    return __expf(-(float)p / TAUF);
}

__global__ __launch_bounds__(256)
void vr_chunks(const float* __restrict__ in, const float* __restrict__ tg,
               float* __restrict__ alpha, float* __restrict__ beta,
               float* __restrict__ evals) {
    // Per-wave LDS tile: 16 segments x 16 elements, padded to stride 20 floats
    // (80B rows -> float4 stores stay 16B aligned; column reads conflict-free).
    __shared__ __align__(16) float lds[8 * 16 * 20];

    const int lane  = threadIdx.x & 31;
    const int wave  = threadIdx.x >> 5;
    const int chunk = blockIdx.x * 8 + wave;   // global chunk id (row*4 + sub)
    const int n     = lane & 15;   // column (segment) index for B/C/D operands
    const int h     = lane >> 4;   // half-wave

    float* wl = &lds[wave * (16 * 20)];

    // --- Constant A operand: K-slices of lower-triangular Toeplitz L -------
    // A (16x4 f32) layout: lane -> M = lane%16 ; VGPR v -> K = v + 2*(lane/16).
    v2f A[4];
#pragma unroll
    for (int k = 0; k < 4; ++k) {
        int c0 = 4 * k + 0 + 2 * h;
        int c1 = 4 * k + 1 + 2 * h;
        A[k].x = (n >= c0) ? dpow(n - c0) : 0.0f;
        A[k].y = (n >= c1) ? dpow(n - c1) : 0.0f;
    }
    // Rank-1 carry fixup coefficients d^(m+1), m = r + 8*h for D VGPR r.
    float dm[8];
#pragma unroll
    for (int r = 0; r < 8; ++r) dm[r] = dpow(r + 1 + 8 * h);
    // Kogge-Stone weights for the 16-segment carry scan (ratio d^16).
    const float w1 = dpow(16), w2 = dpow(32), w4 = dpow(64), w8 = dpow(128);
    const float pc   = dpow(16 * (n + 1));  // applies incoming carry to E_n
    const float d16n = dpow(16 * n);        // beta weight, segment part
    const float w256 = dpow(256);           // beta weight, per-tile decay

    const float4* pin = (const float4*)(in + (size_t)chunk * CHUNK_LEN);
    const float4* ptg = (const float4*)(tg + (size_t)chunk * CHUNK_LEN);

    float carry = 0.0f;   // chunk-local: scan starts from zero state
    float sumsq = 0.0f;   // alpha accumulator
    float bsum  = 0.0f;   // beta  accumulator: sum d^(pos+1) * y_local[pos]
    float bt    = 1.0f;   // d^(256*t)

    for (int t = 0; t < TILES; ++t) {
        const int b4 = t * 64;   // float4 index of tile start
        // Coalesced 512B-per-instruction streams of input & target.
        float4 a0 = pin[b4 + lane];
        float4 a1 = pin[b4 + 32 + lane];
        float4 g0 = ptg[b4 + lane];
        float4 g1 = ptg[b4 + 32 + lane];
        if (t + 1 < TILES) {
            __builtin_prefetch((const void*)&pin[b4 + 64 + lane], 0, 3);
            __builtin_prefetch((const void*)&ptg[b4 + 64 + lane], 0, 3);
        }
        float4 d0 = make_float4(a0.x - g0.x, a0.y - g0.y, a0.z - g0.z, a0.w - g0.w);
        float4 d1 = make_float4(a1.x - g1.x, a1.y - g1.y, a1.z - g1.z, a1.w - g1.w);

        // Stage diff into LDS in (segment, element) layout, padded stride 20.
        // DS ops of one wave execute in order -> compiler barrier suffices.
        const int seg = lane >> 2;
        const int jj  = (lane & 3) << 2;
        asm volatile("" ::: "memory");
        *(float4*)&wl[seg * 20 + jj]       = d0;   // elements  4*lane..+3
        *(float4*)&wl[(seg + 8) * 20 + jj] = d1;   // elements 128+4*lane..+3
        asm volatile("" ::: "memory");

        // B operand (4x16 f32 per K-slice): lane -> N = lane%16,
        // VGPR v -> K-row = v + 2*(lane/16); X[j][n] = diff[n*16 + j].
        v2f B[4];
#pragma unroll
        for (int k = 0; k < 4; ++k) {
            B[k].x = wl[n * 20 + (4 * k + 0 + 2 * h)];
            B[k].y = wl[n * 20 + (4 * k + 1 + 2 * h)];
        }

        // D = L * X : chained 16x16x4 f32 WMMAs over four K-slices.
        v8f acc = {};
        acc = __builtin_amdgcn_wmma_f32_16x16x4_f32(false, A[0], false, B[0],
                                                    (short)0, acc, false, false);
        acc = __builtin_amdgcn_wmma_f32_16x16x4_f32(false, A[1], false, B[1],
                                                    (short)0, acc, false, false);
        acc = __builtin_amdgcn_wmma_f32_16x16x4_f32(false, A[2], false, B[2],
                                                    (short)0, acc, false, false);
        acc = __builtin_amdgcn_wmma_f32_16x16x4_f32(false, A[3], false, B[3],
                                                    (short)0, acc, false, false);

        // Segment-end locals e_n = D[15][n] live in acc[7], lanes 16..31.
        float E = bcastf(acc[7], 16 + n);   // every lane: e_{lane%16}
        // Weighted inclusive scan: E_n = sum_{j<=n} d^(16(n-j)) e_j.
        float tt;
        tt = bcastf(E, (n - 1) & 15); if (n >= 1) E = fmaf(w1, tt, E);
        tt = bcastf(E, (n - 2) & 15); if (n >= 2) E = fmaf(w2, tt, E);
        tt = bcastf(E, (n - 4) & 15); if (n >= 4) E = fmaf(w4, tt, E);
        tt = bcastf(E, (n - 8) & 15); if (n >= 8) E = fmaf(w8, tt, E);
        E = fmaf(pc, carry, E);             // fold in tile-incoming carry

        float cp = bcastf(E, (n - 1) & 15);
        float c  = (n == 0) ? carry : cp;   // carry into segment n
        carry = bcastf(E, 15);              // carry out of this tile (uniform)

        // y[m][n] = D[m][n] + d^(m+1) * c_n ; accumulate alpha (and beta for
        // the first BETA_TILES tiles, after which d^(256t) underflows to 0).
        if (t < BETA_TILES) {
            const float fb = bt * d16n;     // d^(256t + 16n)
#pragma unroll
            for (int r = 0; r < 8; ++r) {
                float y = fmaf(dm[r], c, acc[r]);
                sumsq = fmaf(y, y, sumsq);
                bsum  = fmaf(fb * dm[r], y, bsum);   // d^(pos+1) * y
            }
            bt *= w256;
        } else {
#pragma unroll
            for (int r = 0; r < 8; ++r) {
                float y = fmaf(dm[r], c, acc[r]);
                sumsq = fmaf(y, y, sumsq);
            }
        }
    }

    // Wave-wide butterfly reductions (deterministic).
#pragma unroll
    for (int off = 16; off > 0; off >>= 1) {
        sumsq += bcastf(sumsq, lane ^ off);
        bsum  += bcastf(bsum,  lane ^ off);
    }
    if (lane == 0) {
        alpha[chunk] = sumsq;
        beta[chunk]  = bsum;
        evals[chunk] = carry;   // chunk-local end value E (exact true carry)
    }
}

__global__ __launch_bounds__(256)
void vr_combine(const float* __restrict__ alpha, const float* __restrict__ beta,
                const float* __restrict__ evals, float* __restrict__ out) {
    __shared__ float s[256];
    const int t = threadIdx.x;
    const float d2    = __expf(-2.0f / TAUF);
    const float gamma = d2 / (1.0f - d2);   // sum_{k>=1} d^(2k); tail ~ d^8192 = 0

    float acc = 0.0f;
#pragma unroll
    for (int rr = 0; rr < 4; ++rr) {        // 256 threads x 4 rows = 1024 rows
        const int row = t * 4 + rr;
        float c = 0.0f;                     // carry into first chunk
        float S = 0.0f;
#pragma unroll
        for (int i = 0; i < CHUNKS; ++i) {
            const int idx = row * CHUNKS + i;
            S = fmaf(2.0f * c, beta[idx], S + alpha[idx]);
            S = fmaf(c * c, gamma, S);
            c = evals[idx];                 // exact: d^CHUNK_LEN == 0
        }
        acc += S;
    }
    s[t] = acc;
    __syncthreads();
#pragma unroll
    for (int off = 128; off > 0; off >>= 1) {
        if (t < off) s[t] += s[t + off];
        __syncthreads();
    }
    if (t == 0) out[0] = sqrtf(s[0] * (1.0f / TAUF));
}

extern "C" void kernel_launch(void* const* d_in, const int* in_sizes, int n_in,
                              void* d_out, int out_size, void* d_ws, size_t ws_size,
                              hipStream_t stream) {
    const float* in = (const float*)d_in[0];
    const float* tg = (const float*)d_in[1];
    float* alpha = (float*)d_ws;            // 4096 floats
    float* beta  = alpha + NCHUNK;          // 4096 floats
    float* evals = beta + NCHUNK;           // 4096 floats (48 KB total)

    vr_chunks<<<NCHUNK / 8, 256, 0, stream>>>(in, tg, alpha, beta, evals);
    vr_combine<<<1, 256, 0, stream>>>(alpha, beta, evals, (float*)d_out);
}